// PartialAttentionMasking_35407710388231
// MI455X (gfx1250) — compile-verified
//
#include <hip/hip_runtime.h>

typedef float v2f __attribute__((ext_vector_type(2)));
typedef float v8f __attribute__((ext_vector_type(8)));

#define HW_   50176   // 224*224
#define C_    96
#define B_    16
#define KSEL_ 25088   // 0.5 * HW

// ---------------------------------------------------------------------------
// Kernel 1: per-position channel energy^2 via V_WMMA_F32_16X16X4_F32.
// One wave handles a tile of 16 consecutive positions. For each 4-channel
// chunk, A (16 pos x 4 ch) and B = A^T (4 ch x 16 pos) have identical VGPR
// contents under the f32 WMMA layouts, so one 2-VGPR load feeds both operands.
// D[i][i] accumulates sum_c x[c][pos+i]^2 over all 96 channels.
// ---------------------------------------------------------------------------
__global__ __launch_bounds__(256)
void energy_wmma_kernel(const float* __restrict__ x, float* __restrict__ energy2) {
    const int wave = (blockIdx.x * blockDim.x + threadIdx.x) >> 5;
    const int lane = threadIdx.x & 31;
    const int tiles_per_batch = HW_ / 16;          // 3136
    const int b    = wave / tiles_per_batch;
    const int pos0 = (wave % tiles_per_batch) * 16;

    const int half = lane >> 4;                    // 0: K={0,1}, 1: K={2,3}
    const int col  = lane & 15;                    // position within tile
    const float* xb = x + (size_t)b * C_ * HW_ + pos0 + col;

    v8f acc = {};
#pragma unroll
    for (int c0 = 0; c0 < C_; c0 += 4) {
        const float* p = xb + (size_t)(c0 + 2 * half) * HW_;
        v2f a;
        a.x = p[0];        // channel c0 + 2*half
        a.y = p[HW_];      // channel c0 + 2*half + 1
        // D = A*B + C ; A == B in registers (symmetric layouts)
        acc = __builtin_amdgcn_wmma_f32_16x16x4_f32(
            /*neg_a=*/false, a, /*neg_b=*/false, a,
            /*c_mod=*/(short)0, acc, /*reuse_a=*/false, /*reuse_b=*/false);
    }

    // Diagonal extraction per the 16x16 f32 C/D layout:
    //   M=N=p, p in 0..7  -> lane p,      acc[p]
    //   M=N=p, p in 8..15 -> lane p+16,   acc[p-8]
    int r = 0, p = 0;
    bool active = false;
    if (lane < 8)        { r = lane;      p = lane;      active = true; }
    else if (lane >= 24) { r = lane - 24; p = lane - 16; active = true; }

    float v = acc[0];
    if (r == 1) v = acc[1];
    if (r == 2) v = acc[2];
    if (r == 3) v = acc[3];
    if (r == 4) v = acc[4];
    if (r == 5) v = acc[5];
    if (r == 6) v = acc[6];
    if (r == 7) v = acc[7];

    if (active)
        energy2[(size_t)b * HW_ + pos0 + p] = v;
}

// ---------------------------------------------------------------------------
// Kernel 2: exact k-th largest energy^2 per batch via 4-pass MSB-first radix
// select on fp32 bit patterns (all values >= 0, so uint order == float order).
// One 1024-thread block per batch; data (200 KB/batch) sits in L2.
// ---------------------------------------------------------------------------
__global__ __launch_bounds__(1024)
void radix_select_kernel(const float* __restrict__ energy2,
                         unsigned* __restrict__ thr) {
    __shared__ unsigned hist[256];
    __shared__ unsigned s_prefix;
    __shared__ unsigned s_remaining;

    const int b = blockIdx.x;
    const unsigned* e = (const unsigned*)(energy2 + (size_t)b * HW_);
    const int tid = threadIdx.x;

    if (tid == 0) { s_prefix = 0; s_remaining = KSEL_; }

    for (int pass = 0; pass < 4; ++pass) {
        const int shift = 24 - pass * 8;
        if (tid < 256) hist[tid] = 0;
        __syncthreads();

        const unsigned prefix = s_prefix;
        for (int i = tid; i < HW_; i += 1024) {
            unsigned u = e[i];
            if (pass == 0 || (u >> (shift + 8)) == prefix)
                atomicAdd(&hist[(u >> shift) & 255u], 1u);
        }
        __syncthreads();

        if (tid == 0) {
            unsigned rem = s_remaining, cum = 0;
            int bin = 0;
            for (int j = 255; j >= 0; --j) {
                if (cum + hist[j] >= rem) { bin = j; break; }
                cum += hist[j];
            }
            s_remaining = rem - cum;                 // still needed inside bin
            s_prefix = (prefix << 8) | (unsigned)bin;
        }
        __syncthreads();
    }

    if (tid == 0) thr[b] = s_prefix;                 // bits of k-th largest
}

// ---------------------------------------------------------------------------
// Kernel 3: out = x * mask. Each thread owns 4 consecutive positions; its mask
// is computed once from energy^2 vs threshold, then all 96 channels stream
// through with 128-bit loads/stores (fully coalesced).
// ---------------------------------------------------------------------------
__global__ __launch_bounds__(256)
void apply_mask_kernel(const float* __restrict__ x,
                       const float* __restrict__ energy2,
                       const unsigned* __restrict__ thr,
                       float* __restrict__ out) {
    const int blocks_per_batch = HW_ / 1024;         // 49
    const int b  = blockIdx.x / blocks_per_batch;
    const int p0 = (blockIdx.x % blocks_per_batch) * 1024;
    const int tid = threadIdx.x;

    const unsigned t = thr[b];
    const uint4 ev = ((const uint4*)(energy2 + (size_t)b * HW_ + p0))[tid];
    float4 m;
    m.x = (ev.x >= t) ? 1.0f : 0.0f;
    m.y = (ev.y >= t) ? 1.0f : 0.0f;
    m.z = (ev.z >= t) ? 1.0f : 0.0f;
    m.w = (ev.w >= t) ? 1.0f : 0.0f;

    const float4* xp = (const float4*)(x   + (size_t)b * C_ * HW_ + p0) + tid;
    float4*       op = (float4*)      (out + (size_t)b * C_ * HW_ + p0) + tid;
    const size_t cstride = HW_ / 4;

#pragma unroll 4
    for (int c = 0; c < C_; ++c) {
        float4 xv = xp[c * cstride];
        xv.x *= m.x; xv.y *= m.y; xv.z *= m.z; xv.w *= m.w;
        op[c * cstride] = xv;
    }
}

// ---------------------------------------------------------------------------
extern "C" void kernel_launch(void* const* d_in, const int* in_sizes, int n_in,
                              void* d_out, int out_size, void* d_ws, size_t ws_size,
                              hipStream_t stream) {
    const float* x  = (const float*)d_in[0];
    float* out      = (float*)d_out;
    float* energy2  = (float*)d_ws;
    unsigned* thr   = (unsigned*)((char*)d_ws + (size_t)B_ * HW_ * sizeof(float));

    // 16 batches * 3136 tiles = 50176 waves; 8 waves (256 thr) per block
    const int energy_blocks = (B_ * (HW_ / 16)) / 8;   // 6272
    energy_wmma_kernel<<<energy_blocks, 256, 0, stream>>>(x, energy2);

    radix_select_kernel<<<B_, 1024, 0, stream>>>(energy2, thr);

    const int apply_blocks = B_ * (HW_ / 1024);        // 784
    apply_mask_kernel<<<apply_blocks, 256, 0, stream>>>(x, energy2, thr, out);
}